// ASAPooling_13348758356089
// MI455X (gfx1250) — compile-verified
//
#include <hip/hip_runtime.h>
#include <math.h>

// ---- problem constants (match reference) ----
#define NN 8192          // nodes
#define EE 131072        // edges
#define DD 128           // feature dim
#define KK 4096          // kept nodes (K = ceil(0.5*N))
#define ME (EE + NN)     // edges + self loops
#define NEG_SLOPE 0.2f

typedef __attribute__((ext_vector_type(2))) float v2f;
typedef __attribute__((ext_vector_type(8))) float v8f;

#define WMMA_F32(A, B, ACC) \
  __builtin_amdgcn_wmma_f32_16x16x4_f32(false, (A), false, (B), (short)0, (ACC), false, false)

// row/col of logical edge i (i<EE: real edge, else self loop)
__device__ __forceinline__ void edge_rc(int i, const int* __restrict__ src,
                                        const int* __restrict__ off, int& r, int& c) {
  if (i < EE) { int s = src[i]; r = s; c = (s + 1 + off[i]) % NN; }
  else        { r = i - EE; c = r; }
}

// float atomic max via sign-aware integer punning
__device__ __forceinline__ void atomicMaxFloat(float* addr, float value) {
  if (value >= 0.0f) atomicMax((int*)addr, __float_as_int(value));
  else               atomicMin((unsigned int*)addr, __float_as_uint(value));
}

// ---------------- init ----------------
__global__ void init_small(float* xq, float* smax, float* ssum, float* xnew,
                           float* agg, int* cnt) {
  int i = blockIdx.x * blockDim.x + threadIdx.x;
  if (i < NN * DD) { xq[i] = -INFINITY; xnew[i] = 0.0f; }
  if (i < NN) { smax[i] = -INFINITY; ssum[i] = 0.0f; agg[i] = 0.0f; cnt[i] = 0; }
}

__global__ void zero_f32(float* p, int n) {
  int stride = gridDim.x * blockDim.x;
  for (int i = blockIdx.x * blockDim.x + threadIdx.x; i < n; i += stride) p[i] = 0.0f;
}

// ---------------- stage 1: x_q = segment_max(x[row], col) ----------------
__global__ void segmax_kernel(const float* __restrict__ x, const int* __restrict__ src,
                              const int* __restrict__ off, float* __restrict__ xq) {
  int idx = blockIdx.x * blockDim.x + threadIdx.x;
  if (idx >= ME * DD) return;
  int i = idx >> 7, d = idx & 127;
  int r, c; edge_rc(i, src, off, r, c);
  atomicMaxFloat(&xq[c * DD + d], x[r * DD + d]);
}

// ---------------- stage 2: xqW = x_q @ W_lin + b  (WMMA f32 16x16x4) --------
// 8 waves/block, each wave owns one 16x16 output tile (all 8 column tiles).
__global__ void xqw_wmma_kernel(const float* __restrict__ xq, const float* __restrict__ W,
                                const float* __restrict__ b_lin, float* __restrict__ xqW) {
  int wave = threadIdx.x >> 5;
  int lane = threadIdx.x & 31;
  int half = lane >> 4, l16 = lane & 15;
  int n0 = blockIdx.x * 16;     // node-row tile
  int d0 = wave * 16;           // output-col tile
  v8f acc = {};
#pragma unroll 8
  for (int k0 = 0; k0 < DD; k0 += 4) {
    int ka = k0 + 2 * half;
    v2f a = *(const v2f*)(xq + (n0 + l16) * DD + ka);   // A[m,k], k contiguous
    v2f b;
    b.x = W[ka * DD + d0 + l16];                        // B[k,n]
    b.y = W[(ka + 1) * DD + d0 + l16];
    acc = WMMA_F32(a, b, acc);
  }
  float bias = b_lin[d0 + l16];
#pragma unroll
  for (int v = 0; v < 8; ++v)
    xqW[(n0 + v + 8 * half) * DD + d0 + l16] = acc[v] + bias;
}

// ---------------- stage 3: per-node attention partials ----------------
// u[n] = xqW[n,:] . w_att[:D], v[n] = x[n,:] . w_att[D:]
__global__ void node_uv_kernel(const float* __restrict__ x, const float* __restrict__ xqW,
                               const float* __restrict__ w_att,
                               float* __restrict__ u, float* __restrict__ v) {
  int n = blockIdx.x * blockDim.x + threadIdx.x;
  if (n >= NN) return;
  float su = 0.f, sv = 0.f;
  for (int d = 0; d < DD; ++d) {
    su += xqW[n * DD + d] * w_att[d];
    sv += x[n * DD + d] * w_att[DD + d];
  }
  u[n] = su; v[n] = sv;
}

__global__ void score_smax_kernel(const int* __restrict__ src, const int* __restrict__ off,
                                  const float* __restrict__ u, const float* __restrict__ v,
                                  const float* __restrict__ b_att,
                                  float* __restrict__ score, float* __restrict__ smax) {
  int i = blockIdx.x * blockDim.x + threadIdx.x;
  if (i >= ME) return;
  int r, c; edge_rc(i, src, off, r, c);
  float s = u[c] + v[r] + b_att[0];
  s = (s > 0.f) ? s : NEG_SLOPE * s;       // leaky relu
  score[i] = s;
  atomicMaxFloat(&smax[c], s);
}

__global__ void expsum_kernel(const int* __restrict__ src, const int* __restrict__ off,
                              const float* __restrict__ smax,
                              float* __restrict__ score, float* __restrict__ ssum) {
  int i = blockIdx.x * blockDim.x + threadIdx.x;
  if (i >= ME) return;
  int r, c; edge_rc(i, src, off, r, c);
  float e = expf(score[i] - smax[c]);
  score[i] = e;
  atomicAdd(&ssum[c], e);
}

__global__ void norm_kernel(const int* __restrict__ src, const int* __restrict__ off,
                            const float* __restrict__ ssum, float* __restrict__ score) {
  int i = blockIdx.x * blockDim.x + threadIdx.x;
  if (i >= ME) return;
  int r, c; edge_rc(i, src, off, r, c);
  score[i] = score[i] / ssum[c];
}

// x_new[c,:] += score * x[r,:]
__global__ void xnew_kernel(const float* __restrict__ x, const int* __restrict__ src,
                            const int* __restrict__ off, const float* __restrict__ score,
                            float* __restrict__ xnew) {
  int idx = blockIdx.x * blockDim.x + threadIdx.x;
  if (idx >= ME * DD) return;
  int i = idx >> 7, d = idx & 127;
  int r, c; edge_rc(i, src, off, r, c);
  atomicAdd(&xnew[c * DD + d], x[r * DD + d] * score[i]);
}

// a = x_new@w1+b1, b = x_new@w2, f0 = x_new@w3+b3
__global__ void node_abf_kernel(const float* __restrict__ xnew,
                                const float* __restrict__ w1, const float* __restrict__ b1,
                                const float* __restrict__ w2, const float* __restrict__ w3,
                                const float* __restrict__ b3,
                                float* __restrict__ av, float* __restrict__ bv,
                                float* __restrict__ f0) {
  int n = blockIdx.x * blockDim.x + threadIdx.x;
  if (n >= NN) return;
  float sa = 0.f, sb = 0.f, sf = 0.f;
  for (int d = 0; d < DD; ++d) {
    float xv = xnew[n * DD + d];
    sa += xv * w1[d]; sb += xv * w2[d]; sf += xv * w3[d];
  }
  av[n] = sa + b1[0]; bv[n] = sb; f0[n] = sf + b3[0];
}

__global__ void agg_kernel(const int* __restrict__ src, const int* __restrict__ off,
                           const float* __restrict__ av, const float* __restrict__ bv,
                           float* __restrict__ agg) {
  int i = blockIdx.x * blockDim.x + threadIdx.x;
  if (i >= ME) return;
  int r, c; edge_rc(i, src, off, r, c);
  atomicAdd(&agg[c], av[c] - bv[r]);
}

__global__ void fitness_kernel(const float* __restrict__ f0, const float* __restrict__ agg,
                               float* __restrict__ fit) {
  int n = blockIdx.x * blockDim.x + threadIdx.x;
  if (n >= NN) return;
  float f = f0[n] + agg[n];
  fit[n] = 1.0f / (1.0f + expf(-f));
}

// ---------------- stable descending argsort via exact ranking ----------------
__global__ void rank_kernel(const float* __restrict__ fitness, int* __restrict__ perm_i,
                            int* __restrict__ pos, float* __restrict__ out_perm) {
  __shared__ float lf[NN];             // 32 KB of 320 KB WGP LDS
  for (int t = threadIdx.x; t < NN; t += blockDim.x) lf[t] = fitness[t];
  __syncthreads();
  int n = blockIdx.x * blockDim.x + threadIdx.x;
  if (n >= NN) return;
  float f = lf[n];
  int rank = 0;
  for (int m = 0; m < NN; ++m) {
    float g = lf[m];
    rank += (g > f) || (g == f && m < n);   // stable tie-break by index
  }
  if (rank < KK) { perm_i[rank] = n; out_perm[rank] = (float)n; pos[n] = rank; }
  else pos[n] = -1;
}

__global__ void xout_kernel(const float* __restrict__ xnew, const float* __restrict__ fit,
                            const int* __restrict__ perm_i, float* __restrict__ out_x) {
  int idx = blockIdx.x * blockDim.x + threadIdx.x;
  if (idx >= KK * DD) return;
  int k = idx >> 7, d = idx & 127;
  int n = perm_i[k];
  out_x[idx] = xnew[n * DD + d] * fit[n];
}

// S_perm[r, pos[c]] = score(edge r->c)
__global__ void sperm_kernel(const int* __restrict__ src, const int* __restrict__ off,
                             const int* __restrict__ pos, const float* __restrict__ score,
                             float* __restrict__ outS) {
  int i = blockIdx.x * blockDim.x + threadIdx.x;
  if (i >= ME) return;
  int r, c; edge_rc(i, src, off, r, c);
  int j = pos[c];
  if (j >= 0) outS[r * KK + j] = score[i];
}

// ---------------- CSR of edges keyed by row (out-edges) ----------------
__global__ void deg_kernel(const int* __restrict__ src, const int* __restrict__ off,
                           int* __restrict__ cnt) {
  int i = blockIdx.x * blockDim.x + threadIdx.x;
  if (i >= ME) return;
  int r, c; edge_rc(i, src, off, r, c);
  atomicAdd(&cnt[r], 1);
}

__global__ void scan_kernel(const int* __restrict__ cnt, int* __restrict__ rowstart,
                            int* __restrict__ rowfill) {
  __shared__ int part[256];
  __shared__ int base[256];
  int t = threadIdx.x;
  int s = 0;
  for (int j = 0; j < NN / 256; ++j) s += cnt[t * (NN / 256) + j];
  part[t] = s;
  __syncthreads();
  if (t == 0) {
    int run = 0;
    for (int b = 0; b < 256; ++b) { base[b] = run; run += part[b]; }
  }
  __syncthreads();
  int run = base[t];
  for (int j = 0; j < NN / 256; ++j) {
    int idx = t * (NN / 256) + j;
    rowstart[idx] = run; rowfill[idx] = run;
    run += cnt[idx];
  }
}

__global__ void fill_kernel(const int* __restrict__ src, const int* __restrict__ off,
                            int* __restrict__ rowfill, int* __restrict__ eidx) {
  int i = blockIdx.x * blockDim.x + threadIdx.x;
  if (i >= ME) return;
  int r, c; edge_rc(i, src, off, r, c);
  int slot = atomicAdd(&rowfill[r], 1);
  eidx[slot] = i;
}

// T[r,:] += S_perm[c,:]   using sparsity: row c of S_perm = out-edges of c
__global__ void tbuild_kernel(const int* __restrict__ src, const int* __restrict__ off,
                              const int* __restrict__ rowstart, const int* __restrict__ cnt,
                              const int* __restrict__ eidx, const int* __restrict__ pos,
                              const float* __restrict__ score, float* __restrict__ T) {
  int i = blockIdx.x * blockDim.x + threadIdx.x;
  if (i >= ME) return;
  int r, c; edge_rc(i, src, off, r, c);
  int s0 = rowstart[c], e0 = s0 + cnt[c];
  for (int p = s0; p < e0; ++p) {
    int e2 = eidx[p];
    int r2, c2; edge_rc(e2, src, off, r2, c2);   // r2 == c
    int j = pos[c2];
    if (j >= 0) atomicAdd(&T[r * KK + j], score[e2]);
  }
}

// ---------------- A_new = S_perm^T @ T (f32 WMMA), zero diagonal ----------------
// S, T: NN x KK row-major; C: KK x KK.
// 2x2 register blocking: each wave computes a 32x32 C tile with 4 accumulators,
// so each A/B fragment is reused twice -> 2 loads per WMMA instead of 4
// (arith intensity 8 FLOP/byte from cache). Block = 8 waves = 128x64 C tile.
__global__ void anew_wmma_kernel(const float* __restrict__ S, const float* __restrict__ T,
                                 float* __restrict__ C) {
  int wave = threadIdx.x >> 5;
  int lane = threadIdx.x & 31;
  int half = lane >> 4, l16 = lane & 15;
  int i0 = (blockIdx.x * 4 + (wave & 3)) * 32;   // C row tile (32 wide)
  int j0 = (blockIdx.y * 2 + (wave >> 2)) * 32;  // C col tile (32 wide)
  v8f acc00 = {}, acc01 = {}, acc10 = {}, acc11 = {};
#pragma unroll 4
  for (int k0 = 0; k0 < NN; k0 += 4) {
    int kr = k0 + 2 * half;
    const float* Sr0 = S + kr * KK + i0 + l16;         // row kr
    const float* Sr1 = Sr0 + KK;                       // row kr+1
    const float* Tr0 = T + kr * KK + j0 + l16;
    const float* Tr1 = Tr0 + KK;
    v2f a0, a1, b0, b1;
    a0.x = Sr0[0];  a0.y = Sr1[0];                     // A[m,k] = S[k, i0+m]
    a1.x = Sr0[16]; a1.y = Sr1[16];
    b0.x = Tr0[0];  b0.y = Tr1[0];                     // B[k,n] = T[k, j0+n]
    b1.x = Tr0[16]; b1.y = Tr1[16];
    acc00 = WMMA_F32(a0, b0, acc00);
    acc01 = WMMA_F32(a0, b1, acc01);
    acc10 = WMMA_F32(a1, b0, acc10);
    acc11 = WMMA_F32(a1, b1, acc11);
  }
#pragma unroll
  for (int v = 0; v < 8; ++v) {
    int r0 = i0 + v + 8 * half;        // rows for acc0x
    int r1 = r0 + 16;                  // rows for acc1x
    int c0 = j0 + l16, c1 = c0 + 16;
    C[r0 * KK + c0] = (r0 == c0) ? 0.0f : acc00[v];
    C[r0 * KK + c1] = (r0 == c1) ? 0.0f : acc01[v];
    C[r1 * KK + c0] = (r1 == c0) ? 0.0f : acc10[v];
    C[r1 * KK + c1] = (r1 == c1) ? 0.0f : acc11[v];
  }
}

// ---------------- launcher ----------------
extern "C" void kernel_launch(void* const* d_in, const int* in_sizes, int n_in,
                              void* d_out, int out_size, void* d_ws, size_t ws_size,
                              hipStream_t stream) {
  (void)in_sizes; (void)n_in; (void)out_size; (void)ws_size;
  const float* x     = (const float*)d_in[0];
  const int*   src   = (const int*)d_in[1];
  const int*   off   = (const int*)d_in[2];
  const float* W_lin = (const float*)d_in[3];
  const float* b_lin = (const float*)d_in[4];
  const float* w_att = (const float*)d_in[5];
  const float* b_att = (const float*)d_in[6];
  const float* w1    = (const float*)d_in[7];
  const float* b1    = (const float*)d_in[8];
  const float* w2    = (const float*)d_in[9];
  const float* w3    = (const float*)d_in[10];
  const float* b3    = (const float*)d_in[11];

  float* out   = (float*)d_out;
  float* out_x = out;                          // KK*DD
  float* out_A = out + KK * DD;                // KK*KK
  float* out_p = out_A + (size_t)KK * KK;      // KK
  float* out_S = out_p + KK;                   // NN*KK

  char* w = (char*)d_ws;
  auto alloc = [&](size_t bytes) -> char* {
    char* p = w; w += (bytes + 255) & ~(size_t)255; return p;
  };
  float* xq    = (float*)alloc((size_t)NN * DD * 4);
  float* xqW   = (float*)alloc((size_t)NN * DD * 4);
  float* xnew  = (float*)alloc((size_t)NN * DD * 4);
  float* T     = (float*)alloc((size_t)NN * KK * 4);   // 134 MB scratch
  float* score = (float*)alloc((size_t)ME * 4);
  float* u     = (float*)alloc((size_t)NN * 4);
  float* v     = (float*)alloc((size_t)NN * 4);
  float* av    = (float*)alloc((size_t)NN * 4);
  float* bv    = (float*)alloc((size_t)NN * 4);
  float* f0    = (float*)alloc((size_t)NN * 4);
  float* aggv  = (float*)alloc((size_t)NN * 4);
  float* smax  = (float*)alloc((size_t)NN * 4);
  float* ssum  = (float*)alloc((size_t)NN * 4);
  float* fit   = (float*)alloc((size_t)NN * 4);
  int* cnt     = (int*)alloc((size_t)NN * 4);
  int* rowst   = (int*)alloc((size_t)NN * 4);
  int* rowfill = (int*)alloc((size_t)NN * 4);
  int* eidx    = (int*)alloc((size_t)ME * 4);
  int* perm_i  = (int*)alloc((size_t)KK * 4);
  int* pos     = (int*)alloc((size_t)NN * 4);

  const int TB = 256;
  init_small<<<(NN * DD + TB - 1) / TB, TB, 0, stream>>>(xq, smax, ssum, xnew, aggv, cnt);
  zero_f32<<<4096, TB, 0, stream>>>(T, NN * KK);
  zero_f32<<<4096, TB, 0, stream>>>(out_S, NN * KK);

  segmax_kernel<<<(ME * DD + TB - 1) / TB, TB, 0, stream>>>(x, src, off, xq);
  xqw_wmma_kernel<<<NN / 16, TB, 0, stream>>>(xq, W_lin, b_lin, xqW);
  node_uv_kernel<<<NN / TB, TB, 0, stream>>>(x, xqW, w_att, u, v);
  score_smax_kernel<<<(ME + TB - 1) / TB, TB, 0, stream>>>(src, off, u, v, b_att, score, smax);
  expsum_kernel<<<(ME + TB - 1) / TB, TB, 0, stream>>>(src, off, smax, score, ssum);
  norm_kernel<<<(ME + TB - 1) / TB, TB, 0, stream>>>(src, off, ssum, score);
  xnew_kernel<<<(ME * DD + TB - 1) / TB, TB, 0, stream>>>(x, src, off, score, xnew);
  node_abf_kernel<<<NN / TB, TB, 0, stream>>>(xnew, w1, b1, w2, w3, b3, av, bv, f0);
  agg_kernel<<<(ME + TB - 1) / TB, TB, 0, stream>>>(src, off, av, bv, aggv);
  fitness_kernel<<<NN / TB, TB, 0, stream>>>(f0, aggv, fit);
  rank_kernel<<<NN / TB, TB, 0, stream>>>(fit, perm_i, pos, out_p);
  xout_kernel<<<(KK * DD + TB - 1) / TB, TB, 0, stream>>>(xnew, fit, perm_i, out_x);
  sperm_kernel<<<(ME + TB - 1) / TB, TB, 0, stream>>>(src, off, pos, score, out_S);

  deg_kernel<<<(ME + TB - 1) / TB, TB, 0, stream>>>(src, off, cnt);
  scan_kernel<<<1, 256, 0, stream>>>(cnt, rowst, rowfill);
  fill_kernel<<<(ME + TB - 1) / TB, TB, 0, stream>>>(src, off, rowfill, eidx);
  tbuild_kernel<<<(ME + TB - 1) / TB, TB, 0, stream>>>(src, off, rowst, cnt, eidx, pos, score, T);

  anew_wmma_kernel<<<dim3(KK / 128, KK / 64), TB, 0, stream>>>(out_S, T, out_A);
}